// GraphEncoder_45294725104222
// MI455X (gfx1250) — compile-verified
//
#include <hip/hip_runtime.h>

typedef __attribute__((ext_vector_type(2))) float v2f;
typedef __attribute__((ext_vector_type(8))) float v8f;

// ---------------------------------------------------------------- utilities
__global__ void zero_f32_kernel(float* __restrict__ p, long n) {
  long i = (long)blockIdx.x * blockDim.x + threadIdx.x;
  long stride = (long)gridDim.x * blockDim.x;
  for (; i < n; i += stride) p[i] = 0.0f;
}

__global__ void degree_kernel(const int* __restrict__ src, const int* __restrict__ dst,
                              float* __restrict__ outd, float* __restrict__ ind, int nE) {
  int e = blockIdx.x * blockDim.x + threadIdx.x;
  if (e < nE) {
    atomicAdd(&outd[src[e]], 1.0f);
    atomicAdd(&ind[dst[e]], 1.0f);
  }
}

__global__ void norm_kernel(const float* __restrict__ outd, const float* __restrict__ ind,
                            float* __restrict__ onorm, float* __restrict__ inorm, int n) {
  int i = blockIdx.x * blockDim.x + threadIdx.x;
  if (i < n) {
    onorm[i] = rsqrtf(fmaxf(outd[i], 1.0f));
    inorm[i] = rsqrtf(fmaxf(ind[i], 1.0f));
  }
}

// agg[dst] += h[src] * out_norm[src]; one thread per (edge, 4-float chunk).
// D = 128 fixed -> 32 chunks per edge.
__global__ void aggregate_kernel(const float* __restrict__ h, const float* __restrict__ onorm,
                                 const int* __restrict__ src, const int* __restrict__ dst,
                                 float* __restrict__ agg, int nE) {
  long tid = (long)blockIdx.x * blockDim.x + threadIdx.x;
  int e = (int)(tid >> 5);
  if (e >= nE) return;
  int c = ((int)tid & 31) << 2;
  int s = src[e];
  int d = dst[e];
  float sn = onorm[s];
  const float4 v = *(const float4*)(h + (size_t)s * 128 + c);
  float* o = agg + (size_t)d * 128 + c;
  atomicAdd(o + 0, v.x * sn);
  atomicAdd(o + 1, v.y * sn);
  atomicAdd(o + 2, v.z * sn);
  atomicAdd(o + 3, v.w * sn);
}

// ---------------------------------------------------------------- WMMA GEMM
// out[16-row tile] = epilogue( (agg * in_norm) @ W + bias ), K = 128 fixed.
// MODE 1: relu(x)                         (layer 1)
// MODE 2: relu(0.6*resid + 0.4*x)         (layer 2, resid = h1)
// MODE 3: 0.6*(resid@Wskip + bskip) + 0.4*x   (layer 3, dual WMMA chain)
// One wave per 16x16 output tile; blockDim = 32 * (NCOLS/16).
template <int MODE, int NCOLS>
__global__ __launch_bounds__(32 * (NCOLS / 16))
void gcn_gemm_kernel(const float* __restrict__ agg, const float* __restrict__ inorm,
                     const float* __restrict__ W, const float* __restrict__ bias,
                     const float* __restrict__ resid,
                     const float* __restrict__ Wskip, const float* __restrict__ bskip,
                     float* __restrict__ out, int nrows) {
  constexpr int LDA = 132;  // 128 + 4 pad: lane m hits bank (k + 4m) % 64 -> no conflicts
  __shared__ float As[16 * LDA];
  __shared__ float Hs[(MODE == 3) ? 16 * LDA : 1];

  const int rowBase = blockIdx.x * 16;

  // Cooperative, coalesced fill of the 16x128 A tile (scaled by in_norm).
  for (int idx = threadIdx.x; idx < 16 * 32; idx += blockDim.x) {
    int r = idx >> 5;
    int c = (idx & 31) << 2;
    int gr = rowBase + r;
    float4 v = make_float4(0.f, 0.f, 0.f, 0.f);
    float sc = 0.f;
    if (gr < nrows) {
      v = *(const float4*)(agg + (size_t)gr * 128 + c);
      sc = inorm[gr];
    }
    float* a = &As[r * LDA + c];
    a[0] = v.x * sc; a[1] = v.y * sc; a[2] = v.z * sc; a[3] = v.w * sc;
    if constexpr (MODE == 3) {
      float4 w2 = make_float4(0.f, 0.f, 0.f, 0.f);
      if (gr < nrows) w2 = *(const float4*)(resid + (size_t)gr * 128 + c);
      float* hq = &Hs[r * LDA + c];
      hq[0] = w2.x; hq[1] = w2.y; hq[2] = w2.z; hq[3] = w2.w;
    }
  }
  __syncthreads();

  const int lane = threadIdx.x & 31;
  const int wave = threadIdx.x >> 5;
  const int t = lane >> 4;    // half-wave select
  const int mn = lane & 15;   // A row (m) / B-C-D column (n)
  const int colBase = wave * 16;

  v8f acc = {};
  v8f accS = {};
#pragma unroll 4
  for (int k = 0; k < 128; k += 4) {
    // A 16x4 f32 fragment: VGPR v holds K = 2*t + v for row m (ISA 7.12.2)
    v2f a, b;
    a.x = As[mn * LDA + k + 2 * t];
    a.y = As[mn * LDA + k + 2 * t + 1];
    // B 4x16 fragment: VGPR v holds row K = 2*t + v, column n
    b.x = W[(size_t)(k + 2 * t) * NCOLS + colBase + mn];
    b.y = W[(size_t)(k + 2 * t + 1) * NCOLS + colBase + mn];
    acc = __builtin_amdgcn_wmma_f32_16x16x4_f32(false, a, false, b, (short)0, acc,
                                                false, false);
    if constexpr (MODE == 3) {
      v2f a2, b2;
      a2.x = Hs[mn * LDA + k + 2 * t];
      a2.y = Hs[mn * LDA + k + 2 * t + 1];
      b2.x = Wskip[(size_t)(k + 2 * t) * NCOLS + colBase + mn];
      b2.y = Wskip[(size_t)(k + 2 * t + 1) * NCOLS + colBase + mn];
      accS = __builtin_amdgcn_wmma_f32_16x16x4_f32(false, a2, false, b2, (short)0,
                                                   accS, false, false);
    }
  }

  // C/D 16x16 f32 layout: VGPR r -> row (r + 8*t), column n
  const int col = colBase + mn;
  const float bv = bias[col];
  const float bsv = (MODE == 3) ? bskip[col] : 0.0f;

  if (rowBase + 16 <= nrows) {
    // Full tile (always the case when nrows % 16 == 0): branchless epilogue,
    // per-row stores are 64B-contiguous per half-wave.
#pragma unroll
    for (int r = 0; r < 8; ++r) {
      int row = rowBase + r + 8 * t;
      float v = acc[r] + bv;
      if constexpr (MODE == 1) {
        v = fmaxf(v, 0.0f);
      } else if constexpr (MODE == 2) {
        v = fmaxf(0.6f * resid[(size_t)row * NCOLS + col] + 0.4f * v, 0.0f);
      } else {
        v = 0.6f * (accS[r] + bsv) + 0.4f * v;
      }
      out[(size_t)row * NCOLS + col] = v;
    }
  } else {
#pragma unroll
    for (int r = 0; r < 8; ++r) {
      int row = rowBase + r + 8 * t;
      if (row < nrows) {
        float v = acc[r] + bv;
        if constexpr (MODE == 1) {
          v = fmaxf(v, 0.0f);
        } else if constexpr (MODE == 2) {
          v = fmaxf(0.6f * resid[(size_t)row * NCOLS + col] + 0.4f * v, 0.0f);
        } else {
          v = 0.6f * (accS[r] + bsv) + 0.4f * v;
        }
        out[(size_t)row * NCOLS + col] = v;
      }
    }
  }
}

// ---------------------------------------------------------------- launcher
extern "C" void kernel_launch(void* const* d_in, const int* in_sizes, int n_in,
                              void* d_out, int out_size, void* d_ws, size_t ws_size,
                              hipStream_t stream) {
  const float* feats = (const float*)d_in[0];
  const int* eidx    = (const int*)d_in[1];
  const float* W1  = (const float*)d_in[2];
  const float* b1  = (const float*)d_in[3];
  const float* W2  = (const float*)d_in[4];
  const float* b2  = (const float*)d_in[5];
  const float* W3  = (const float*)d_in[6];
  const float* b3  = (const float*)d_in[7];
  const float* Wsk = (const float*)d_in[8];
  const float* bsk = (const float*)d_in[9];

  const int n  = in_sizes[0] / 128;  // N nodes
  const int nE = in_sizes[1] / 2;    // E edges
  const int* src = eidx;
  const int* dst = eidx + nE;

  float* ws    = (float*)d_ws;
  float* outd  = ws;
  float* ind   = outd + n;
  float* onorm = ind + n;
  float* inorm = onorm + n;
  float* agg   = inorm + n;
  float* h1    = agg + (size_t)n * 128;
  float* h2    = h1 + (size_t)n * 128;
  float* outp  = (float*)d_out;

  const int rowTiles = (n + 15) / 16;
  const long aggThreads = (long)nE * 32;
  const int aggBlocks = (int)((aggThreads + 255) / 256);

  // degrees + norms
  zero_f32_kernel<<<256, 256, 0, stream>>>(outd, 2L * n);
  degree_kernel<<<(nE + 255) / 256, 256, 0, stream>>>(src, dst, outd, ind, nE);
  norm_kernel<<<(n + 255) / 256, 256, 0, stream>>>(outd, ind, onorm, inorm, n);

  // layer 1: h1 = relu(Agg(feats) @ W1 + b1)
  zero_f32_kernel<<<512, 256, 0, stream>>>(agg, (long)n * 128);
  aggregate_kernel<<<aggBlocks, 256, 0, stream>>>(feats, onorm, src, dst, agg, nE);
  gcn_gemm_kernel<1, 128><<<rowTiles, 256, 0, stream>>>(
      agg, inorm, W1, b1, nullptr, nullptr, nullptr, h1, n);

  // layer 2: h2 = relu(0.6*h1 + 0.4*(Agg(h1) @ W2 + b2))
  zero_f32_kernel<<<512, 256, 0, stream>>>(agg, (long)n * 128);
  aggregate_kernel<<<aggBlocks, 256, 0, stream>>>(h1, onorm, src, dst, agg, nE);
  gcn_gemm_kernel<2, 128><<<rowTiles, 256, 0, stream>>>(
      agg, inorm, W2, b2, h1, nullptr, nullptr, h2, n);

  // layer 3: out = 0.6*(h2 @ skip_W + skip_b) + 0.4*(Agg(h2) @ W3 + b3)
  zero_f32_kernel<<<512, 256, 0, stream>>>(agg, (long)n * 128);
  aggregate_kernel<<<aggBlocks, 256, 0, stream>>>(h2, onorm, src, dst, agg, nE);
  gcn_gemm_kernel<3, 64><<<rowTiles, 128, 0, stream>>>(
      agg, inorm, W3, b3, h2, Wsk, bsk, outp, n);
}